// GroupPolicy_2740189135245
// MI455X (gfx1250) — compile-verified
//
#include <hip/hip_runtime.h>
#include <hip/hip_bf16.h>

#define F_DIM 128
#define H_DIM 128
#define H2_DIM 64
#define T_DIM 8
#define G_DIM 64

typedef __attribute__((ext_vector_type(2))) float v2f;
typedef __attribute__((ext_vector_type(8))) float v8f;

// ---------------------------------------------------------------------------
// init: deg (stored in dinv buffer) = 1.0 (self loop), pool/cnt zeroed
// ---------------------------------------------------------------------------
__global__ void init_kernel(float* __restrict__ deg, float* __restrict__ pool,
                            float* __restrict__ cnt, int N) {
    int i = blockIdx.x * blockDim.x + threadIdx.x;
    if (i < N) deg[i] = 1.0f;
    if (i < G_DIM * H_DIM) pool[i] = 0.0f;
    if (i < G_DIM) cnt[i] = 0.0f;
}

// deg[dst] += 1 per edge
__global__ void degree_kernel(const int* __restrict__ dst, float* __restrict__ deg, int E) {
    int e = blockIdx.x * blockDim.x + threadIdx.x;
    if (e < E) atomicAdd(&deg[dst[e]], 1.0f);
}

// dinv = deg^-1/2 (in place)
__global__ void dinv_kernel(float* __restrict__ deg, int N) {
    int i = blockIdx.x * blockDim.x + threadIdx.x;
    if (i < N) deg[i] = rsqrtf(deg[i]);
}

// ---------------------------------------------------------------------------
// h0 = x @ W via V_WMMA_F32_16X16X4_F32.
// Block = 8 waves, all sharing one 16-column tile of W staged in LDS (8 KB).
// Wave w handles row tile (m_super*8 + w). Inner loop is branch-free:
//   A (16x4 f32): a[v] = x[row0+sub][k0+2*half+v]   (one float2 global load)
//   B (4x16 f32): b[v] = lw[(k0+2*half+v)*16 + sub] (ds_load_2addr, no bank conflict)
//   D (16x16 f32): acc[v] -> h0[row0+8*half+v][col0+sub]
// Store epilogue: wave-uniform bounds test -> unguarded store clause on the
// fast path; per-row guards only for a partial last tile.
// ---------------------------------------------------------------------------
__global__ void gemm_wmma_kernel(const float* __restrict__ x, const float* __restrict__ W,
                                 float* __restrict__ h0, int N, int m_tiles) {
    __shared__ float lw[F_DIM * 16];            // W[:, col0:col0+16]
    const int tid  = threadIdx.x;
    const int wid  = tid >> 5;
    const int lane = tid & 31;
    const int col_tile = blockIdx.x & 7;        // n_tiles = H/16 = 8
    const int m_super  = blockIdx.x >> 3;
    const int col0 = col_tile * 16;

    // Cooperative stage of the W column slice: 512 float4 over 256 threads.
    #pragma unroll
    for (int it = 0; it < 2; ++it) {
        const int i  = tid + it * 256;          // 0..511
        const int k  = i >> 2;                  // 0..127
        const int c4 = (i & 3) * 4;             // 0,4,8,12
        const float4 w4 = *(const float4*)(W + (size_t)k * H_DIM + col0 + c4);
        *(float4*)(lw + k * 16 + c4) = w4;
    }
    __syncthreads();

    const int mt = m_super * 8 + wid;
    if (mt >= m_tiles) return;                  // whole-wave exit after barrier

    const int row0 = mt * 16;
    const int half = lane >> 4;
    const int sub  = lane & 15;
    int r = row0 + sub;
    if (r > N - 1) r = N - 1;                   // clamp loads; stores handled below
    const float* __restrict__ xrow = x + (size_t)r * F_DIM;

    v8f acc = {};
    #pragma unroll 8
    for (int k0 = 0; k0 < F_DIM; k0 += 4) {
        const int ka = k0 + 2 * half;
        const float2 t = *(const float2*)(xrow + ka);
        v2f a;  a.x = t.x;           a.y = t.y;
        v2f bm; bm.x = lw[ka * 16 + sub]; bm.y = lw[(ka + 1) * 16 + sub];
        acc = __builtin_amdgcn_wmma_f32_16x16x4_f32(
            /*neg_a=*/false, a, /*neg_b=*/false, bm,
            /*c_mod=*/(short)0, acc, /*reuse_a=*/false, /*reuse_b=*/false);
    }

    float* __restrict__ orow = h0 + (size_t)(row0 + half * 8) * H_DIM + col0 + sub;
    if (row0 + 15 < N) {                        // wave-uniform: full tile in bounds
        #pragma unroll
        for (int v = 0; v < 8; ++v) orow[(size_t)v * H_DIM] = acc[v];
    } else {                                    // partial last tile only
        #pragma unroll
        for (int v = 0; v < 8; ++v) {
            if (row0 + half * 8 + v < N) orow[(size_t)v * H_DIM] = acc[v];
        }
    }
}

// agg = h0 * dinv^2  (self-loop message; also serves as agg initialization)
__global__ void selfloop_kernel(const float* __restrict__ h0, const float* __restrict__ dinv,
                                float* __restrict__ agg, int total) {
    int i = blockIdx.x * blockDim.x + threadIdx.x;
    if (i < total) {
        float dv = dinv[i >> 7];           // i / H_DIM
        agg[i] = h0[i] * dv * dv;
    }
}

// Edge scatter: one wave per edge; lane handles 4 contiguous floats.
// h0 and agg are L2-resident (25.6 MB each << 192 MB L2).
__global__ void scatter_kernel(const int* __restrict__ src, const int* __restrict__ dst,
                               const float* __restrict__ dinv, const float* __restrict__ h0,
                               float* __restrict__ agg, int E) {
    const int wave = (int)((blockIdx.x * blockDim.x + threadIdx.x) >> 5);
    const int lane = threadIdx.x & 31;
    if (wave >= E) return;
    const int s = src[wave];
    const int d = dst[wave];
    const float norm = dinv[s] * dinv[d];
    const float4 v = *(const float4*)(h0 + (size_t)s * H_DIM + lane * 4);
    float* a = agg + (size_t)d * H_DIM + lane * 4;
    atomicAdd(a + 0, v.x * norm);
    atomicAdd(a + 1, v.y * norm);
    atomicAdd(a + 2, v.z * norm);
    atomicAdd(a + 3, v.w * norm);
}

// relu(agg + b) -> mean-pool accumulation. One wave per node.
__global__ void relu_pool_kernel(const float* __restrict__ agg, const float* __restrict__ b,
                                 const int* __restrict__ batch, float* __restrict__ pool,
                                 float* __restrict__ cnt, int N) {
    const int wave = (int)((blockIdx.x * blockDim.x + threadIdx.x) >> 5);
    const int lane = threadIdx.x & 31;
    if (wave >= N) return;
    const int g = batch[wave];
    const float4 v  = *(const float4*)(agg + (size_t)wave * H_DIM + lane * 4);
    const float4 bb = *(const float4*)(b + lane * 4);
    float* p = pool + (size_t)g * H_DIM + lane * 4;
    atomicAdd(p + 0, fmaxf(v.x + bb.x, 0.0f));
    atomicAdd(p + 1, fmaxf(v.y + bb.y, 0.0f));
    atomicAdd(p + 2, fmaxf(v.z + bb.z, 0.0f));
    atomicAdd(p + 3, fmaxf(v.w + bb.w, 0.0f));
    if (lane == 0) atomicAdd(&cnt[g], 1.0f);
}

// Head: mean, fc1+relu, actor softmax, critic. Single block (work is tiny).
__global__ void head_kernel(const float* __restrict__ pool, const float* __restrict__ cnt,
                            const float* __restrict__ fc1_w, const float* __restrict__ fc1_b,
                            const float* __restrict__ actor_w, const float* __restrict__ actor_b,
                            const float* __restrict__ critic_w, const float* __restrict__ critic_b,
                            float* __restrict__ out) {
    __shared__ float gs[G_DIM * H_DIM];     // 32 KB
    __shared__ float z[G_DIM * H2_DIM];     // 16 KB
    __shared__ float logit[G_DIM * T_DIM];  // 2 KB
    const int tid = threadIdx.x;

    for (int i = tid; i < G_DIM * H_DIM; i += blockDim.x) {
        int g = i / H_DIM;
        gs[i] = pool[i] / fmaxf(cnt[g], 1.0f);
    }
    __syncthreads();

    for (int i = tid; i < G_DIM * H2_DIM; i += blockDim.x) {
        int g = i / H2_DIM, j = i % H2_DIM;
        float s = fc1_b[j];
        #pragma unroll 8
        for (int k = 0; k < H_DIM; ++k) s += gs[g * H_DIM + k] * fc1_w[k * H2_DIM + j];
        z[i] = fmaxf(s, 0.0f);
    }
    __syncthreads();

    for (int i = tid; i < G_DIM * T_DIM; i += blockDim.x) {
        int g = i / T_DIM, t = i % T_DIM;
        float s = actor_b[t];
        #pragma unroll 8
        for (int k = 0; k < H2_DIM; ++k) s += z[g * H2_DIM + k] * actor_w[k * T_DIM + t];
        logit[i] = s;
    }
    __syncthreads();

    if (tid < G_DIM) {
        const int g = tid;
        float m = -1e30f;
        #pragma unroll
        for (int t = 0; t < T_DIM; ++t) m = fmaxf(m, logit[g * T_DIM + t]);
        float e[T_DIM], ssum = 0.0f;
        #pragma unroll
        for (int t = 0; t < T_DIM; ++t) { e[t] = expf(logit[g * T_DIM + t] - m); ssum += e[t]; }
        const float inv = 1.0f / ssum;
        #pragma unroll
        for (int t = 0; t < T_DIM; ++t) out[g * T_DIM + t] = e[t] * inv;

        float v = critic_b[0];
        #pragma unroll 8
        for (int k = 0; k < H2_DIM; ++k) v += z[g * H2_DIM + k] * critic_w[k];
        out[G_DIM * T_DIM + g] = v;
    }
}

extern "C" void kernel_launch(void* const* d_in, const int* in_sizes, int n_in,
                              void* d_out, int out_size, void* d_ws, size_t ws_size,
                              hipStream_t stream) {
    const float* x        = (const float*)d_in[0];
    const int*   ei       = (const int*)  d_in[1];
    const int*   batch    = (const int*)  d_in[2];
    const float* W        = (const float*)d_in[3];
    const float* b        = (const float*)d_in[4];
    const float* fc1_w    = (const float*)d_in[5];
    const float* fc1_b    = (const float*)d_in[6];
    const float* actor_w  = (const float*)d_in[7];
    const float* actor_b  = (const float*)d_in[8];
    const float* critic_w = (const float*)d_in[9];
    const float* critic_b = (const float*)d_in[10];
    float* out = (float*)d_out;

    const int N = in_sizes[2];
    const int E = in_sizes[1] / 2;
    const int* src = ei;
    const int* dst = ei + E;

    // Workspace layout (floats)
    float* ws   = (float*)d_ws;
    float* dinv = ws;                                  // N       (deg, then dinv)
    float* h0   = dinv + N;                            // N*H
    float* agg  = h0 + (size_t)N * H_DIM;              // N*H
    float* pool = agg + (size_t)N * H_DIM;             // G*H
    float* cnt  = pool + (size_t)G_DIM * H_DIM;        // G

    const int TB = 256;

    init_kernel<<<(N + TB - 1) / TB, TB, 0, stream>>>(dinv, pool, cnt, N);
    degree_kernel<<<(E + TB - 1) / TB, TB, 0, stream>>>(dst, dinv, E);
    dinv_kernel<<<(N + TB - 1) / TB, TB, 0, stream>>>(dinv, N);

    const int m_tiles = (N + 15) / 16;
    const int gemm_blocks = ((m_tiles + 7) / 8) * 8;   // (m_super groups) x (8 col tiles)
    gemm_wmma_kernel<<<gemm_blocks, TB, 0, stream>>>(x, W, h0, N, m_tiles);

    const int totNH = N * H_DIM;
    selfloop_kernel<<<(totNH + TB - 1) / TB, TB, 0, stream>>>(h0, dinv, agg, totNH);

    scatter_kernel<<<(E + 7) / 8, TB, 0, stream>>>(src, dst, dinv, h0, agg, E);

    relu_pool_kernel<<<(N + 7) / 8, TB, 0, stream>>>(agg, b, batch, pool, cnt, N);

    head_kernel<<<1, TB, 0, stream>>>(pool, cnt, fc1_w, fc1_b, actor_w, actor_b,
                                      critic_w, critic_b, out);
}